// GaussianBlur_19988777796258
// MI455X (gfx1250) — compile-verified
//
#include <hip/hip_runtime.h>

typedef __attribute__((ext_vector_type(2))) float v2f;
typedef __attribute__((ext_vector_type(8))) float v8f;
typedef __attribute__((address_space(1))) int gint;   // global int
typedef __attribute__((address_space(3))) int sint;   // LDS int

#define RAD   4
#define TAPS  9
#define TILE  32              // 32x32 output tile per workgroup
#define IN_H  (TILE + 2*RAD)  // 40 input rows (with vertical halo)
#define IN_W  48              // 3 x 16 columns (covers horizontal halo, 40 used)
#define MID_S 50              // mid stride (even -> keeps float2 LDS loads 8B aligned)

// Normalized 9-tap Gaussian, sigma=1, truncate=4 (matches skimage defaults).
__device__ __constant__ float GW[TAPS] = {
    1.3383067e-04f, 4.4318615e-03f, 5.3991128e-02f, 2.4197144e-01f,
    3.9894347e-01f,
    2.4197144e-01f, 5.3991128e-02f, 4.4318615e-03f, 1.3383067e-04f,
};

// Fused separable Gaussian blur via fp32 WMMA banded matmuls:
//   mid = Wv(16x24) * X(24x16)   (vertical pass,  A = band, B = data)
//   out = mid(16x24) * Wh(24x16) (horizontal pass, A = data, B = band)
// Band element for lane l, K-slot k is w[k - (l&15)] in BOTH operand layouts,
// so one register array Wq[] serves both passes.
__global__ __launch_bounds__(128) void gauss_blur_wmma(
    const float* __restrict__ x, float* __restrict__ out)
{
    __shared__ float in_t[IN_H][IN_W];   // edge-clamped input tile
    __shared__ float mid[TILE][MID_S];   // vertical-pass result (48 cols used)

    const int H = 512, W = 512;
    const int tid  = threadIdx.x;
    const int lane = tid & 31;
    // Wave id made provably wave-uniform -> scalar branches, so WMMA always
    // executes with EXEC == all ones (ISA requirement).
    const int wvu  = __builtin_amdgcn_readfirstlane(tid >> 5);   // 0..3
    const int hi   = lane >> 4;      // lane half (K split)
    const int ml   = lane & 15;      // M index (A) / N index (B)

    const int tilesx = W / TILE;     // 16
    const int tx = blockIdx.x % tilesx;
    const int ty = blockIdx.x / tilesx;
    const int b  = blockIdx.y;
    const int r0 = ty * TILE;
    const int c0 = tx * TILE;

    // ---- stage 1: edge-clamped tile load (1920 = 15*128, no divergence) ----
    const float* img = x + (size_t)b * H * W;
    for (int idx = tid; idx < IN_H * IN_W; idx += 128) {
        int i  = idx / IN_W, j = idx % IN_W;
        int rr = min(max(r0 - RAD + i, 0), H - 1);
        int cc = min(max(c0 - RAD + j, 0), W - 1);
#if __has_builtin(__builtin_amdgcn_global_load_async_to_lds_b32)
        // CDNA5 async LDS-DMA: per-lane clamped global addr -> per-lane LDS
        // addr, tracked with ASYNCcnt (no VGPR staging).
        __builtin_amdgcn_global_load_async_to_lds_b32(
            (gint*)(img + rr * W + cc), (sint*)&in_t[i][j], 0, 0);
#else
        in_t[i][j] = img[rr * W + cc];
#endif
    }

    // Per-lane banded-Gaussian operand registers, K = 24 in 6 chunks of 4
    // (overlaps with the in-flight async loads).
    v2f Wq[6];
#pragma unroll
    for (int q = 0; q < 6; ++q) {
        int k0 = 4 * q + 2 * hi;
        int d0 = k0 - ml, d1 = k0 + 1 - ml;
        Wq[q].x = (d0 >= 0 && d0 < TAPS) ? GW[d0] : 0.0f;
        Wq[q].y = (d1 >= 0 && d1 < TAPS) ? GW[d1] : 0.0f;
    }

#if __has_builtin(__builtin_amdgcn_global_load_async_to_lds_b32)
    // Drain this wave's async LDS writes before the workgroup barrier.
    asm volatile("s_wait_asynccnt 0x0" ::: "memory");
#endif
    __syncthreads();

    // ---- stage 2: vertical pass, 6 tiles (2 row-tiles x 3 col-tiles) ----
    for (int tv = wvu; tv < 6; tv += 4) {   // scalar-uniform loop
        int rt = tv / 3, ct = tv % 3;
        v8f acc = {};
#pragma unroll
        for (int q = 0; q < 6; ++q) {
            int kk = 4 * q + 2 * hi;        // input row offset within band
            v2f bb;
            bb.x = in_t[rt * 16 + kk    ][ct * 16 + ml];
            bb.y = in_t[rt * 16 + kk + 1][ct * 16 + ml];
            acc = __builtin_amdgcn_wmma_f32_16x16x4_f32(
                false, Wq[q], false, bb, (short)0, acc, false, false);
        }
        // D layout: VGPR g -> row g (lanes 0-15) / g+8 (lanes 16-31)
#pragma unroll
        for (int g = 0; g < 8; ++g)
            mid[rt * 16 + g + 8 * hi][ct * 16 + ml] = acc[g];
    }

    __syncthreads();

    // ---- stage 3: horizontal pass, one 16x16 output tile per wave ----
    {
        int rt = wvu >> 1, ct = wvu & 1;    // scalar-uniform
        int rr = rt * 16 + ml;              // A-operand row (lane = M)
        int cb = ct * 16;
        v8f acc = {};
#pragma unroll
        for (int q = 0; q < 6; ++q) {
            int kk = 4 * q + 2 * hi;
            // consecutive cols, even index + even stride -> 8B-aligned b64 load
            v2f aa = *(const v2f*)&mid[rr][cb + kk];
            acc = __builtin_amdgcn_wmma_f32_16x16x4_f32(
                false, aa, false, Wq[q], (short)0, acc, false, false);
        }
        float* orow = out + ((size_t)b * H + r0 + rt * 16 + 8 * hi) * W
                          + (c0 + cb + ml);
#pragma unroll
        for (int g = 0; g < 8; ++g)
            orow[(size_t)g * W] = acc[g];
    }
}

extern "C" void kernel_launch(void* const* d_in, const int* in_sizes, int n_in,
                              void* d_out, int out_size, void* d_ws, size_t ws_size,
                              hipStream_t stream) {
    const float* x = (const float*)d_in[0];
    float* out     = (float*)d_out;
    const int Bn   = in_sizes[0] / (512 * 512);   // 128 batches
    dim3 grid((512 / TILE) * (512 / TILE), Bn);   // (256, 128)
    gauss_blur_wmma<<<grid, 128, 0, stream>>>(x, out);
    (void)n_in; (void)out_size; (void)d_ws; (void)ws_size;
}